// MambaBlock_87239375716403
// MI455X (gfx1250) — compile-verified
//
#include <hip/hip_runtime.h>
#include <hip/hip_bf16.h>

// ---------------- problem constants ----------------
constexpr int Bn = 4;          // batch
constexpr int Ln = 2048;       // sequence length
constexpr int DM = 1024;       // d_model
constexpr int DS = 64;         // d_state
constexpr int DI = 2048;       // d_inner
constexpr int RT = Bn * Ln;    // 8192 flattened rows

typedef float v2f __attribute__((ext_vector_type(2)));
typedef float v8f __attribute__((ext_vector_type(8)));

// D = A(16x4,f32) * B(4x16,f32) + C(16x16,f32)   (V_WMMA_F32_16X16X4_F32)
__device__ __forceinline__ v8f wmma4(v2f a, v2f b, v8f c) {
    return __builtin_amdgcn_wmma_f32_16x16x4_f32(
        false, a, false, b, (short)0, c, false, false);
}

__device__ __forceinline__ v8f vzero8() {
    v8f z = {0.f, 0.f, 0.f, 0.f, 0.f, 0.f, 0.f, 0.f};
    return z;
}

// ---- CDNA5 async global->LDS copy, 16B per lane (GLOBAL_LOAD_ASYNC_TO_LDS_B128)
// VDST operand = LDS byte address (low 32 bits of the generic shared pointer).
__device__ __forceinline__ void async_copy16(const float* g, const float* l) {
    const unsigned lds_addr = (unsigned)(uintptr_t)l;
    asm volatile("global_load_async_to_lds_b128 %0, %1, off"
                 :: "v"(lds_addr), "v"(g)
                 : "memory");
}

template <int N>
__device__ __forceinline__ void wait_async() {
#if __has_builtin(__builtin_amdgcn_s_wait_asynccnt)
    __builtin_amdgcn_s_wait_asynccnt((short)N);
#else
    asm volatile("s_wait_asynccnt %0" :: "i"(N) : "memory");
#endif
}

// ---------------------------------------------------------------------------
// K0: fold output projections:  Mc_t[d, s] = sum_i W_out[d,i] * D_mat[s,i]
//     [1024 x 64], K = 2048.  One 16x16 tile per wave; 256 tiles total.
// ---------------------------------------------------------------------------
__global__ void __launch_bounds__(256) k_fold(const float* __restrict__ W_out,
                                              const float* __restrict__ D_mat,
                                              float* __restrict__ Mc_t) {
    const int lane   = threadIdx.x & 31;
    const int waveId = blockIdx.x * 8 + (threadIdx.x >> 5);
    const int ts = waveId & 3;            // DS/16 = 4
    const int td = waveId >> 2;           // DM/16 = 64
    const int d0 = td * 16, s0 = ts * 16;
    const int mr = lane & 15;
    const int kb = (lane >> 4) << 1;

    const float* ap = W_out + (size_t)(d0 + mr) * DI + kb;  // A[m][k]
    const float* bp = D_mat + (size_t)(s0 + mr) * DI + kb;  // B[k][n] = D_mat[n][k]
    v8f acc = vzero8();
    for (int k = 0; k < DI; k += 4) {
        v2f a = *(const v2f*)(ap + k);
        v2f b = *(const v2f*)(bp + k);
        acc = wmma4(a, b, acc);
    }
    const int rb = d0 + ((lane >> 4) << 3);
#pragma unroll
    for (int j = 0; j < 8; ++j)
        Mc_t[(size_t)(rb + j) * DS + s0 + mr] = acc[j];
}

// ---------------------------------------------------------------------------
// K1: h1 = x @ W_in^T + b_in    [8192 x 2048], K = 1024
//     Block-tiled: 128x64 macro tile per 8-wave block; each wave owns a
//     32x32 sub-tile (2x2 WMMA accumulators).  A/B K-chunks (BK=64) staged
//     in LDS via double-buffered async global->LDS B128 copies.
//     LDS row stride 68 floats: 16B-aligned B128 stores + conflict-free
//     ds_load_b64 (4-bank stride between consecutive rows).
// ---------------------------------------------------------------------------
constexpr int BM = 128, BNt = 64, BK = 64, LDP = 68;

__global__ void __launch_bounds__(256) k_in_proj(const float* __restrict__ x,
                                                 const float* __restrict__ W_in,
                                                 const float* __restrict__ b_in,
                                                 float* __restrict__ h1) {
    __shared__ __align__(16) float As[2][BM][LDP];    // ~68KB
    __shared__ __align__(16) float Bs[2][BNt][LDP];   // ~34KB

    const int tid  = threadIdx.x;
    const int lane = tid & 31;
    const int w    = tid >> 5;
    const int wr   = w & 3;               // row quadrant  (32 rows)
    const int wc   = w >> 2;              // column half   (32 cols)
    const int tm   = blockIdx.x >> 5;     // RT/BM  = 64 row blocks
    const int tn   = blockIdx.x & 31;     // DI/BNt = 32 col blocks
    const int r0   = tm * BM;
    const int c0   = tn * BNt;
    const int mr   = lane & 15;
    const int kb   = (lane >> 4) << 1;

    // ---- async stage of one K-chunk into LDS buffer `buf` ----
    auto stage = [&](int buf, int kc) {
        // A tile: 128 rows x 64 cols = 2048 16B-quads; 8 per thread
#pragma unroll
        for (int i = 0; i < 8; ++i) {
            const int q   = i * 256 + tid;
            const int row = q >> 4;
            const int c4  = (q & 15) << 2;
            async_copy16(x + (size_t)(r0 + row) * DM + kc + c4,
                         &As[buf][row][c4]);
        }
        // B tile: 64 rows x 64 cols = 1024 quads; 4 per thread
#pragma unroll
        for (int i = 0; i < 4; ++i) {
            const int q   = i * 256 + tid;
            const int row = q >> 4;
            const int c4  = (q & 15) << 2;
            async_copy16(W_in + (size_t)(c0 + row) * DM + kc + c4,
                         &Bs[buf][row][c4]);
        }
    };

    v8f acc00 = vzero8(), acc01 = vzero8(), acc10 = vzero8(), acc11 = vzero8();

    stage(0, 0);
    constexpr int NK = DM / BK;           // 16 chunks
    for (int kci = 0; kci < NK; ++kci) {
        const int buf = kci & 1;
        if (kci + 1 < NK) {
            stage(buf ^ 1, (kci + 1) * BK);
            wait_async<12>();             // 12 newer asyncs outstanding -> current chunk done
        } else {
            wait_async<0>();
        }
        __syncthreads();
        const int ar0 = wr * 32 + mr, ar1 = ar0 + 16;
        const int br0 = wc * 32 + mr, br1 = br0 + 16;
#pragma unroll
        for (int kk = 0; kk < BK; kk += 4) {
            v2f a0 = *(const v2f*)&As[buf][ar0][kk + kb];
            v2f a1 = *(const v2f*)&As[buf][ar1][kk + kb];
            v2f b0 = *(const v2f*)&Bs[buf][br0][kk + kb];
            v2f b1 = *(const v2f*)&Bs[buf][br1][kk + kb];
            acc00 = wmma4(a0, b0, acc00);
            acc01 = wmma4(a0, b1, acc01);
            acc10 = wmma4(a1, b0, acc10);
            acc11 = wmma4(a1, b1, acc11);
        }
        __syncthreads();                  // all waves done before buffer reuse
    }

    // ---- epilogue: bias + store ----
    const int col0 = c0 + wc * 32 + mr;
    const int col1 = col0 + 16;
    const float bias0 = b_in[col0];
    const float bias1 = b_in[col1];
    const int rbase = r0 + wr * 32 + ((lane >> 4) << 3);
#pragma unroll
    for (int j = 0; j < 8; ++j) {
        h1[(size_t)(rbase + j) * DI + col0]      = acc00[j] + bias0;
        h1[(size_t)(rbase + j) * DI + col1]      = acc01[j] + bias1;
        h1[(size_t)(rbase + 16 + j) * DI + col0] = acc10[j] + bias0;
        h1[(size_t)(rbase + 16 + j) * DI + col1] = acc11[j] + bias1;
    }
}

// ---------------------------------------------------------------------------
// K2: fused depthwise causal conv (K=4) -> LDS tile -> WMMA GEMMs with
//     W_x (waves 0-3) and W_dt (waves 4-7) -> softplus/exp ->
//     dA, dBx written in scan layout [B*DS, L].
// ---------------------------------------------------------------------------
__global__ void __launch_bounds__(256) k_ssm_param(
        const float* __restrict__ h1,
        const float* __restrict__ conv_w, const float* __restrict__ conv_b,
        const float* __restrict__ W_x,  const float* __restrict__ b_x,
        const float* __restrict__ W_dt, const float* __restrict__ b_dt,
        const float* __restrict__ A_log,
        float* __restrict__ dA_t, float* __restrict__ dBx_t) {
    __shared__ __align__(16) float convA[16][68];
    __shared__ __align__(16) float xs[16][64];
    __shared__ __align__(16) float dts[16][64];

    const int tid  = threadIdx.x;
    const int lane = tid & 31;
    const int w    = tid >> 5;
    const int bIdx = blockIdx.x >> 7;               // Ln/16 = 128 tiles per batch
    const int l0   = (blockIdx.x & 127) * 16;
    const int rowBase = bIdx * Ln;

    const int matSel = w >> 2;                      // 0: x_ssm, 1: dt
    const int s0 = (w & 3) * 16;
    const int mr = lane & 15;
    const int kb = (lane >> 4) << 1;
    const float* Wsel = matSel ? W_dt : W_x;
    const float* bp = Wsel + (size_t)(s0 + mr) * DI + kb;   // B[k][n] = W[n][k]

    v8f acc = vzero8();
    for (int kc = 0; kc < DI; kc += 64) {
        __syncthreads();
        for (int e = tid; e < 16 * 64; e += 256) {
            const int ll = e >> 6, cc = e & 63;
            const int c  = kc + cc;
            float v = conv_b[c];
            const int lb = l0 + ll - 3;
#pragma unroll
            for (int j = 0; j < 4; ++j) {
                const int lg = lb + j;
                if (lg >= 0)
                    v += conv_w[c * 4 + j] * h1[(size_t)(rowBase + lg) * DI + c];
            }
            convA[ll][cc] = v;
        }
        __syncthreads();
#pragma unroll 4
        for (int kk = 0; kk < 64; kk += 4) {
            v2f a = *(const v2f*)&convA[mr][kk + kb];    // ds_load_b64
            v2f b = *(const v2f*)(bp + kc + kk);
            acc = wmma4(a, b, acc);
        }
    }
    const float bias = (matSel ? b_dt : b_x)[s0 + mr];
    const int rb = (lane >> 4) << 3;
    __syncthreads();
#pragma unroll
    for (int j = 0; j < 8; ++j) {
        if (matSel) dts[rb + j][s0 + mr] = acc[j] + bias;
        else        xs [rb + j][s0 + mr] = acc[j] + bias;
    }
    __syncthreads();
    for (int e = tid; e < 16 * 64; e += 256) {
        const int ll = e >> 6, s = e & 63;
        const float raw = dts[ll][s];
        const float sp  = raw > 30.f ? raw : log1pf(expf(raw));   // softplus
        const float As  = -expf(A_log[s]);
        const float dAv = expf(As * sp);
        const float dbx = ((1.f - dAv) / As) * xs[ll][s];
        const size_t idx = (size_t)(bIdx * DS + s) * Ln + (l0 + ll);
        dA_t[idx]  = dAv;
        dBx_t[idx] = dbx;
    }
}

// ---------------------------------------------------------------------------
// K3: recurrence z_l = dA_l * z_{l-1} + dBx_l per (b, s); one wave each.
// ---------------------------------------------------------------------------
__global__ void __launch_bounds__(32) k_scan(const float* __restrict__ dA_t,
                                             const float* __restrict__ dBx_t,
                                             float* __restrict__ z) {
    const int row  = blockIdx.x;          // b*DS + s
    const int lane = threadIdx.x;
    const int b = row >> 6, s = row & 63;
    const float* ap = dA_t  + (size_t)row * Ln + lane * 64;
    const float* dp = dBx_t + (size_t)row * Ln + lane * 64;
    __builtin_prefetch(ap, 0, 1);         // global_prefetch_b8
    __builtin_prefetch(dp, 0, 1);

    float A = 1.f, Bc = 0.f;
    for (int t = 0; t < 64; ++t) {
        const float a = ap[t], d = dp[t];
        A  = a * A;
        Bc = a * Bc + d;
    }
    for (int off = 1; off < 32; off <<= 1) {
        const float Au = __shfl_up(A,  off, 32);
        const float Bu = __shfl_up(Bc, off, 32);
        if (lane >= off) { Bc = A * Bu + Bc; A = A * Au; }
    }
    float zstart = __shfl_up(Bc, 1, 32);
    if (lane == 0) zstart = 0.f;
    float zv = zstart;
    const int lbase = b * Ln + lane * 64;
    for (int t = 0; t < 64; ++t) {
        zv = ap[t] * zv + dp[t];
        z[(size_t)(lbase + t) * DS + s] = zv;
    }
}

// ---------------------------------------------------------------------------
// K4: out = z @ Mc_t^T + b_out     [8192 x 1024], K = 64
// ---------------------------------------------------------------------------
__global__ void __launch_bounds__(256) k_out(const float* __restrict__ z,
                                             const float* __restrict__ Mc_t,
                                             const float* __restrict__ b_out,
                                             float* __restrict__ out) {
    const int lane   = threadIdx.x & 31;
    const int waveId = blockIdx.x * 8 + (threadIdx.x >> 5);
    const int tn = waveId & 63;           // DM/16 = 64
    const int tm = waveId >> 6;           // RT/16 = 512
    const int r0 = tm * 16, d0 = tn * 16;
    const int mr = lane & 15;
    const int kb = (lane >> 4) << 1;

    const float* ap = z    + (size_t)(r0 + mr) * DS + kb;
    const float* bp = Mc_t + (size_t)(d0 + mr) * DS + kb;
    v8f acc = vzero8();
#pragma unroll
    for (int k = 0; k < DS; k += 4) {
        v2f a = *(const v2f*)(ap + k);
        v2f b = *(const v2f*)(bp + k);
        acc = wmma4(a, b, acc);
    }
    const float bias = b_out[d0 + mr];
    const int rb = r0 + ((lane >> 4) << 3);
#pragma unroll
    for (int j = 0; j < 8; ++j)
        out[(size_t)(rb + j) * DM + d0 + mr] = acc[j] + bias;
}

// ---------------------------------------------------------------------------
extern "C" void kernel_launch(void* const* d_in, const int* in_sizes, int n_in,
                              void* d_out, int out_size, void* d_ws, size_t ws_size,
                              hipStream_t stream) {
    const float* x      = (const float*)d_in[0];
    const float* W_in   = (const float*)d_in[1];
    const float* b_in   = (const float*)d_in[2];
    const float* conv_w = (const float*)d_in[3];
    const float* conv_b = (const float*)d_in[4];
    const float* W_x    = (const float*)d_in[5];
    const float* b_x    = (const float*)d_in[6];
    const float* W_dt   = (const float*)d_in[7];
    const float* b_dt   = (const float*)d_in[8];
    const float* A_log  = (const float*)d_in[9];
    const float* D_mat  = (const float*)d_in[10];
    const float* W_out  = (const float*)d_in[11];
    const float* b_out  = (const float*)d_in[12];
    float* out = (float*)d_out;

    float* h1    = (float*)d_ws;                       // 8192 * 2048
    float* dA_t  = h1    + (size_t)RT * DI;            // 256 * 2048
    float* dBx_t = dA_t  + (size_t)Bn * DS * Ln;       // 256 * 2048
    float* z     = dBx_t + (size_t)Bn * DS * Ln;       // 8192 * 64
    float* Mc_t  = z     + (size_t)RT * DS;            // 1024 * 64

    k_fold<<<32, 256, 0, stream>>>(W_out, D_mat, Mc_t);
    k_in_proj<<<(RT / BM) * (DI / BNt), 256, 0, stream>>>(x, W_in, b_in, h1);
    k_ssm_param<<<Bn * (Ln / 16), 256, 0, stream>>>(h1, conv_w, conv_b,
                                                    W_x, b_x, W_dt, b_dt, A_log,
                                                    dA_t, dBx_t);
    k_scan<<<Bn * DS, 32, 0, stream>>>(dA_t, dBx_t, z);
    k_out<<<(RT / 16) * (DM / 16) / 8, 256, 0, stream>>>(z, Mc_t, b_out, out);
}